// FSCLoss_87058987090062
// MI455X (gfx1250) — compile-verified
//
#include <hip/hip_runtime.h>
#include <hip/hip_bf16.h>

// ---------------------------------------------------------------------------
// FSC loss on MI455X (gfx1250, wave32).
// 3D FFT implemented as three DFT-by-GEMM stages using V_WMMA_F32_16X16X4_F32
// (full-precision f32 matrix path -- fp16/bf16 WMMA would wreck FFT accuracy).
// Per-batch working set (~48 MB) lives in the 192 MB L2.
// W matrix is pre-swizzled into float4 quads (re_n, re_n+1, im_n, im_n+1) so
// each lane fetches its whole W operand pair with ONE global_load_b128/K-step.
// ---------------------------------------------------------------------------

typedef __attribute__((ext_vector_type(2))) float v2f;
typedef __attribute__((ext_vector_type(8))) float v8f;

#define FSC_N    128
#define FSC_VOL  (1 << 21)            // 128^3
#define FSC_BC   8                    // B*C
#define FSC_S    65                   // shells 0..64
#define ACC_PER_BC (FSC_S * 3)        // num / out_power / tgt_power

static __device__ __forceinline__ v8f wmma4(v2f a, v2f b, v8f c) {
  // D = A(16x4 f32) * B(4x16 f32) + C(16x16 f32)
  return __builtin_amdgcn_wmma_f32_16x16x4_f32(
      /*neg_a=*/false, a, /*neg_b=*/false, b,
      /*c_mod=*/(short)0, c, /*reuse_a=*/false, /*reuse_b=*/false);
}

// ---------------------------------------------------------------------------
// Orthonormal forward DFT matrix, swizzled: entry (np, col) holds
//   { Wre[2np][col], Wre[2np+1][col], Wim[2np][col], Wim[2np+1][col] }
// with W[n][k] = e^{-2*pi*i*n*k/128} / sqrt(128).
// Angles reduced exactly via (n*k mod 128) -> bit-deterministic table.
// ---------------------------------------------------------------------------
__global__ void dft_matrix_init(float4* __restrict__ Wpk) {
  int idx = blockIdx.x * blockDim.x + threadIdx.x;     // 64*128 entries
  if (idx >= 64 * FSC_N) return;
  int np = idx >> 7, col = idx & 127;
  const float step  = 6.283185307179586f / 128.0f;
  const float scale = 0.08838834764831845f;            // 1/sqrt(128)
  int t0 = ((2 * np)     * col) & 127;                 // exact periodic reduction
  int t1 = ((2 * np + 1) * col) & 127;
  float s0, c0, s1, c1;
  __sincosf((float)t0 * step, &s0, &c0);
  __sincosf((float)t1 * step, &s1, &c1);
  float4 w;
  w.x = c0 * scale;  w.y = c1 * scale;                 // re_n, re_{n+1}
  w.z = -s0 * scale; w.w = -s1 * scale;                // im_n, im_{n+1}
  Wpk[idx] = w;
}

// ---------------------------------------------------------------------------
// One DFT stage as a complex GEMM over 16384 lines x 128 freqs, K = 128.
// Element address of line l, axis-index n:  (l>>7)*SH + (l&127)*SL + n*SN
//   stage X: SH=16384 SL=128 SN=1      (lines = (z,y))   UNIT_SN -> b64 loads
//   stage Y: SH=16384 SL=1   SN=128    (lines = (z,x))
//   stage Z: SH=128   SL=1   SN=16384  (lines = (y,x))
// Output of each stage uses the SAME triple (freq index replaces n).
// DATA_IS_B: data feeds the B operand (lanes = unit-stride x) -> coalesced
// loads and stores for the Y/Z stages.  8 waves per block, 1 C-tile per wave.
// ---------------------------------------------------------------------------
template <bool DATA_IS_B, bool IN_REAL, bool UNIT_SN>
__global__ __launch_bounds__(256, 2)
void fft_stage(const float* __restrict__ inRe, const float* __restrict__ inIm,
               const float4* __restrict__ Wpk,
               float* __restrict__ outRe, float* __restrict__ outIm,
               int SH, int SL, int SN) {
  const int wave = threadIdx.x >> 5;
  const int lane = threadIdx.x & 31;
  const int tile = blockIdx.x * 8 + wave;      // 8192 tiles total
  const int tL = (tile >> 3) << 4;             // line-tile base (16384 lines)
  const int tF = (tile & 7) << 4;              // freq-tile base (128 freqs)
  const int hl  = lane >> 4;                   // lane half (K-pair select)
  const int l15 = lane & 15;

  v8f accRe1 = {}; v8f accRe2 = {};            // re*re , im*im
  v8f accIm1 = {}; v8f accIm2 = {};            // re*im , im*re

  // Swizzled-W per-lane base: entry ((k0>>1) + hl)*128 + (tF + l15).
  const int wlane = hl * FSC_N + tF + l15;
  // Data per-lane base: line dl picked by l15 in both orientations
  // (A rows / B columns both map the 16-line tile onto l15).
  const int dl    = tL + l15;
  const int dbase = (dl >> 7) * SH + (dl & 127) * SL;

#pragma unroll 4
  for (int k0 = 0; k0 < 128; k0 += 4) {
    // one b128: (re_n, re_{n+1}, im_n, im_{n+1}) for this lane's K-pair
    float4 w = Wpk[(k0 >> 1) * FSC_N + wlane];
    v2f wre, wim;
    wre.x = w.x;  wre.y = w.y;
    wim.x = w.z;  wim.y = w.w;

    const int a0 = dbase + (k0 + hl * 2) * SN;
    v2f dre, dim = {};
    if (UNIT_SN) {                              // adjacent -> b64 load
      dre = *(const v2f*)(inRe + a0);
      if (!IN_REAL) dim = *(const v2f*)(inIm + a0);
    } else {
      dre.x = inRe[a0];  dre.y = inRe[a0 + SN];
      if (!IN_REAL) { dim.x = inIm[a0];  dim.y = inIm[a0 + SN]; }
    }

    if (DATA_IS_B) {                 // C[k][l] += W^T-tile x data
      accRe1 = wmma4(wre, dre, accRe1);
      if (!IN_REAL) {
        accRe2 = wmma4(wim, dim, accRe2);
        accIm1 = wmma4(wre, dim, accIm1);
      }
      accIm2 = wmma4(wim, dre, accIm2);
    } else {                         // C[l][k] += data x W-tile
      accRe1 = wmma4(dre, wre, accRe1);
      if (!IN_REAL) {
        accRe2 = wmma4(dim, wim, accRe2);
        accIm2 = wmma4(dim, wre, accIm2);
      }
      accIm1 = wmma4(dre, wim, accIm1);
    }
  }

  v8f cre = IN_REAL ? accRe1 : (accRe1 - accRe2);
  v8f cim;
  if (IN_REAL) cim = DATA_IS_B ? accIm2 : accIm1;
  else         cim = accIm1 + accIm2;

#pragma unroll
  for (int r = 0; r < 8; ++r) {
    int m = r + hl * 8;                        // C row (VGPR index mapping)
    int addr;
    if (DATA_IS_B) {                           // rows = freq, lanes = line
      int kf = tF + m;
      int l  = tL + l15;
      addr = (l >> 7) * SH + (l & 127) * SL + kf * SN;
    } else {                                   // rows = line, lanes = freq
      int l  = tL + m;
      int kf = tF + l15;
      addr = (l >> 7) * SH + (l & 127) * SL + kf * SN;
    }
    outRe[addr] = cre[r];
    outIm[addr] = cim[r];
  }
}

// ---------------------------------------------------------------------------
// Per-point shell scatter: fftshift folded into centered coordinates,
// LDS-first reduction (65 shells x 3 sums) then global atomics.
// ---------------------------------------------------------------------------
__global__ __launch_bounds__(256)
void fsc_accum(const float* __restrict__ mRe, const float* __restrict__ mIm,
               const float* __restrict__ tRe, const float* __restrict__ tIm,
               float* __restrict__ acc, int bc) {
  __shared__ float sacc[ACC_PER_BC];
  const int tid = threadIdx.x;
  if (tid < ACC_PER_BC) sacc[tid] = 0.0f;
  __syncthreads();

  const int idx = blockIdx.x * 256 + tid;      // < 2^21
  const int x = idx & 127, y = (idx >> 7) & 127, z = idx >> 14;
  const int cx = (x < 64) ? x : x - 128;
  const int cy = (y < 64) ? y : y - 128;
  const int cz = (z < 64) ? z : z - 128;
  const int r2 = cx * cx + cy * cy + cz * cz;
  const int shell = (int)sqrtf((float)r2);     // exact trunc for r2 <= 12288

  if (shell <= 64) {
    float mr = mRe[idx], mi = mIm[idx];
    float tr = tRe[idx], ti = tIm[idx];
    atomicAdd(&sacc[shell * 3 + 0], mr * tr + mi * ti);   // Re(out*conj(tgt))
    atomicAdd(&sacc[shell * 3 + 1], mr * mr + mi * mi);   // |out|^2
    atomicAdd(&sacc[shell * 3 + 2], tr * tr + ti * ti);   // |tgt|^2
  }
  __syncthreads();
  if (tid < ACC_PER_BC) atomicAdd(&acc[bc * ACC_PER_BC + tid], sacc[tid]);
}

// ---------------------------------------------------------------------------
// Final scalar: shells 1..64 are exactly the valid shells (all non-empty),
// denom = 64; loss_bc = 1 - sum_s fsc / 64; output = mean over bc.
// ---------------------------------------------------------------------------
__global__ void fsc_final(const float* __restrict__ acc, float* __restrict__ out) {
  __shared__ float ssum[FSC_BC];
  const int t = threadIdx.x;                   // 64 threads, shell = t+1
  if (t < FSC_BC) ssum[t] = 0.0f;
  __syncthreads();
  const int s = t + 1;
#pragma unroll
  for (int bc = 0; bc < FSC_BC; ++bc) {
    float num = acc[bc * ACC_PER_BC + s * 3 + 0];
    float mp  = acc[bc * ACC_PER_BC + s * 3 + 1];
    float tp  = acc[bc * ACC_PER_BC + s * 3 + 2];
    float fsc = num / sqrtf(mp * tp + 1e-6f);
    fsc = fminf(1.0f, fmaxf(-1.0f, fsc));
    atomicAdd(&ssum[bc], fsc);
  }
  __syncthreads();
  if (t == 0) {
    float loss = 0.0f;
#pragma unroll
    for (int bc = 0; bc < FSC_BC; ++bc) loss += 1.0f - ssum[bc] * (1.0f / 64.0f);
    out[0] = loss * (1.0f / (float)FSC_BC);
  }
}

// ---------------------------------------------------------------------------
extern "C" void kernel_launch(void* const* d_in, const int* in_sizes, int n_in,
                              void* d_out, int out_size, void* d_ws, size_t ws_size,
                              hipStream_t stream) {
  (void)in_sizes; (void)n_in; (void)out_size; (void)ws_size;
  const float* model  = (const float*)d_in[0];   // (8,1,128,128,128) f32
  const float* target = (const float*)d_in[1];

  // Workspace layout (floats). Total: 65536 + 6*2^21 ~= 48.3 MB.
  float* base = (float*)d_ws;
  float4* Wpk = (float4*)base;        // 64*128 float4 = 32768 floats
  float* acc  = base + 32768;         // 8*195 = 1560
  float* P0re = base + 65536;
  float* P0im = P0re + FSC_VOL;
  float* P1re = P0im + FSC_VOL;
  float* P1im = P1re + FSC_VOL;
  float* Mre  = P1im + FSC_VOL;
  float* Mim  = Mre  + FSC_VOL;

  dft_matrix_init<<<32, 256, 0, stream>>>(Wpk);
  hipMemsetAsync(acc, 0, FSC_BC * ACC_PER_BC * sizeof(float), stream);

  const int GEMM_BLOCKS = 1024;       // 8192 tiles / 8 waves per block

  for (int b = 0; b < FSC_BC; ++b) {
    for (int v = 0; v < 2; ++v) {
      const float* src = (v ? target : model) + (size_t)b * FSC_VOL;
      float* dRe = v ? P0re : Mre;    // target's final FFT reuses P0
      float* dIm = v ? P0im : Mim;
      // stage X (real input, data as A, unit contract stride):
      fft_stage<false, true, true><<<GEMM_BLOCKS, 256, 0, stream>>>(
          src, nullptr, Wpk, P0re, P0im, 16384, 128, 1);
      // stage Y (complex, data as B):     SH=16384 SL=1  SN=128
      fft_stage<true, false, false><<<GEMM_BLOCKS, 256, 0, stream>>>(
          P0re, P0im, Wpk, P1re, P1im, 16384, 1, 128);
      // stage Z (complex, data as B):     SH=128   SL=1  SN=16384
      fft_stage<true, false, false><<<GEMM_BLOCKS, 256, 0, stream>>>(
          P1re, P1im, Wpk, dRe, dIm, 128, 1, 16384);
    }
    fsc_accum<<<FSC_VOL / 256, 256, 0, stream>>>(Mre, Mim, P0re, P0im, acc, b);
  }
  fsc_final<<<1, 64, 0, stream>>>(acc, (float*)d_out);
}